// Renderer_79147657330969
// MI455X (gfx1250) — compile-verified
//
#include <hip/hip_runtime.h>
#include <stdint.h>

// ---------------------------------------------------------------------------
// NeRF-style volume render + loss, MI455X (gfx1250, wave32).
// One wave per ray (128 samples, 4/lane). Double-buffered async-to-LDS DMA
// (ASYNCcnt) streams rgb+sigma; dists input is dead in the reference and is
// never read (saves ~20% of HBM traffic).
// ---------------------------------------------------------------------------

#define NSAMP 128
#define RPW 8                 // rays per wave (pipeline depth along the loop)
#define WAVES_PER_BLOCK 8
#define RAYS_PER_BLOCK (RPW * WAVES_PER_BLOCK)   // 64 rays / block

#define RGB_BYTES  (NSAMP * 3 * 4)   // 1536
#define SIG_BYTES  (NSAMP * 4)       // 512
#define RAY_BYTES  (RGB_BYTES + SIG_BYTES)  // 2048, double buffered per wave

// ---- CDNA5 async DMA: global -> LDS, tracked by ASYNCcnt -------------------
__device__ __forceinline__ void async_b128(uint32_t lds_addr, uint64_t gaddr) {
  asm volatile("global_load_async_to_lds_b128 %0, %1, off"
               :: "v"(lds_addr), "v"(gaddr) : "memory");
}

template <int N>
__device__ __forceinline__ void wait_asynccnt() {
  asm volatile("s_wait_asynccnt %0" :: "i"(N) : "memory");
}

__device__ __forceinline__ void wait_dscnt0() {
  asm volatile("s_wait_dscnt 0x0" ::: "memory");
}

// ---- wave32 reductions -----------------------------------------------------
__device__ __forceinline__ float wave_sum(float v) {
#pragma unroll
  for (int m = 16; m >= 1; m >>= 1) v += __shfl_xor(v, m, 32);
  return v;
}
__device__ __forceinline__ float wave_max(float v) {
#pragma unroll
  for (int m = 16; m >= 1; m >>= 1) v = fmaxf(v, __shfl_xor(v, m, 32));
  return v;
}

__global__ __launch_bounds__(256) void nerf_render_loss_kernel(
    const float* __restrict__ raw_rgb,    // [N,128,3]
    const float* __restrict__ raw_sigma,  // [N,128]
    const float* __restrict__ targets,    // [N,3]
    float* __restrict__ out,              // [N*3] rgb_map ++ [N] losses
    int nrays) {
  __shared__ __align__(16) unsigned char smem[WAVES_PER_BLOCK * 2 * RAY_BYTES];

  const int lane = threadIdx.x & 31;
  const int wave = threadIdx.x >> 5;
  const int wave_global = blockIdx.x * WAVES_PER_BLOCK + wave;
  const int ray0 = wave_global * RPW;

  unsigned char* wbase = smem + wave * (2 * RAY_BYTES);
  // low 32 bits of a generic shared pointer == wave-relative LDS offset
  const uint32_t lds_base = (uint32_t)(uintptr_t)(void*)wbase;

  auto issue = [&](int ray, int buf) {
    const char* grgb = (const char*)raw_rgb + (size_t)ray * RGB_BYTES;
    const char* gsig = (const char*)raw_sigma + (size_t)ray * SIG_BYTES;
    const uint32_t l = lds_base + (uint32_t)buf * RAY_BYTES + (uint32_t)lane * 16;
    // 4 x b128 per wave = 2048 B (rgb 1536 + sigma 512), fully coalesced
    async_b128(l + 0u,          (uint64_t)(uintptr_t)(grgb + lane * 16 + 0));
    async_b128(l + 512u,        (uint64_t)(uintptr_t)(grgb + lane * 16 + 512));
    async_b128(l + 1024u,       (uint64_t)(uintptr_t)(grgb + lane * 16 + 1024));
    async_b128(l + RGB_BYTES,   (uint64_t)(uintptr_t)(gsig + lane * 16));
  };

  if (ray0 < nrays) issue(ray0, 0);

  for (int i = 0; i < RPW; ++i) {
    const int ray = ray0 + i;
    if (ray >= nrays) break;

    // Ensure prior ds reads of the buffer we are about to overwrite finished,
    // then prefetch ray i+1 while we compute ray i.
    if (i + 1 < RPW && ray + 1 < nrays) {
      wait_dscnt0();
      issue(ray + 1, (i + 1) & 1);
      wait_asynccnt<4>();   // batch for ray i complete (in-order), next in flight
    } else {
      wait_asynccnt<0>();
    }

    const unsigned char* buf = wbase + (size_t)(i & 1) * RAY_BYTES;
    const float4* rgb4 = (const float4*)buf;               // 96 x float4
    const float4* sig4 = (const float4*)(buf + RGB_BYTES); // 32 x float4

    const float4 sg = sig4[lane];
    const float4 ra = rgb4[lane * 3 + 0];
    const float4 rb = rgb4[lane * 3 + 1];
    const float4 rc = rgb4[lane * 3 + 2];

    const float sig[4] = {sg.x, sg.y, sg.z, sg.w};
    const float rr[4][3] = {{ra.x, ra.y, ra.z},
                            {ra.w, rb.x, rb.y},
                            {rb.z, rb.w, rc.x},
                            {rc.y, rc.z, rc.w}};

    // lp = -softplus(sigma); lnp = sigma - softplus(sigma)  (one softplus!)
    float lp[4], lnp[4];
#pragma unroll
    for (int j = 0; j < 4; ++j) {
      const float x = sig[j];
      const float sp = fmaxf(x, 0.0f) + __logf(1.0f + __expf(-fabsf(x)));
      lp[j] = -sp;
      lnp[j] = x - sp;
    }

    // exclusive scan of lp along 128 samples: local prefix + wave scan
    const float p0 = lp[0];
    const float p1 = p0 + lp[1];
    const float p2 = p1 + lp[2];
    const float p3 = p2 + lp[3];
    const float e[4] = {0.0f, p0, p1, p2};

    float incl = p3;
#pragma unroll
    for (int off = 1; off < 32; off <<= 1) {
      const float t = __shfl_up(incl, off, 32);
      if (lane >= off) incl += t;
    }
    const float excl = incl - p3;  // sum of lp over all samples before this lane

    const float* tp = targets + (size_t)ray * 3;
    const float t0 = tp[0], t1 = tp[1], t2 = tp[2];

    float acc0 = 0.0f, acc1 = 0.0f, acc2 = 0.0f;
    float logp[4];
    float mloc = -3.0e38f;
#pragma unroll
    for (int j = 0; j < 4; ++j) {
      const float lw = lnp[j] + excl + e[j];       // log_weights[s]
      const float w = __expf(lw);
      const float c0 = 1.0f / (1.0f + __expf(-rr[j][0]));
      const float c1 = 1.0f / (1.0f + __expf(-rr[j][1]));
      const float c2 = 1.0f / (1.0f + __expf(-rr[j][2]));
      acc0 += w * c0; acc1 += w * c1; acc2 += w * c2;
      const float d0 = c0 - t0, d1 = c1 - t1, d2 = c2 - t2;
      logp[j] = lw - 0.5f * (d0 * d0 + d1 * d1 + d2 * d2);
      mloc = fmaxf(mloc, logp[j]);
    }

    // 129th (background) element lives on lane 31: log_w[S] = sum lp[0..126]
    float logp_bg = 0.0f;
    const bool has_bg = (lane == 31);
    if (has_bg) {
      const float lw_last = excl + p2;
      logp_bg = lw_last - 0.5f * (t0 * t0 + t1 * t1 + t2 * t2);
      mloc = fmaxf(mloc, logp_bg);
    }

    const float m = wave_max(mloc);
    float sA = 0.0f, sB = 0.0f;
#pragma unroll
    for (int j = 0; j < 4; ++j) {
      const float ev = __expf(logp[j] - m);
      sA += ev;
      sB += ev * logp[j];
    }
    if (has_bg) {
      const float ev = __expf(logp_bg - m);
      sA += ev;
      sB += ev * logp_bg;
    }
    sA = wave_sum(sA);
    sB = wave_sum(sB);
    acc0 = wave_sum(acc0);
    acc1 = wave_sum(acc1);
    acc2 = wave_sum(acc2);

    if (lane == 0) {
      out[(size_t)ray * 3 + 0] = acc0;
      out[(size_t)ray * 3 + 1] = acc1;
      out[(size_t)ray * 3 + 2] = acc2;
      out[(size_t)nrays * 3 + ray] = -sB / sA;   // losses
    }
  }
}

extern "C" void kernel_launch(void* const* d_in, const int* in_sizes, int n_in,
                              void* d_out, int out_size, void* d_ws, size_t ws_size,
                              hipStream_t stream) {
  const float* raw_rgb = (const float*)d_in[0];
  const float* raw_sigma = (const float*)d_in[1];
  // d_in[2] = dists: computed into `sigma` in the original but never used.
  const float* targets = (const float*)d_in[3];
  float* out = (float*)d_out;

  const int nrays = in_sizes[1] / NSAMP;
  const int blocks = (nrays + RAYS_PER_BLOCK - 1) / RAYS_PER_BLOCK;
  nerf_render_loss_kernel<<<blocks, WAVES_PER_BLOCK * 32, 0, stream>>>(
      raw_rgb, raw_sigma, targets, out, nrays);
}